// ACT_R_42820823941698
// MI455X (gfx1250) — compile-verified
//
#include <hip/hip_runtime.h>
#include <stdint.h>

// ACT-R forward on MI455X (gfx1250).
//   sp : [128, 16384] f32 (row-major, batch contiguous)
//   w  : [5] f32 = (a, c, s, tau, h)
//   out: [127, 16384] f32
//
// One batch column per thread. Column data staged into LDS once via the CDNA5
// async global->LDS DMA path (GLOBAL_LOAD_ASYNC_TO_LDS_B32 + s_wait_asynccnt),
// then the O(S^2) recurrence runs entirely out of LDS.
// Per-pair cost after the exp(m)=S algebraic simplification:
//   1x ds_load_b64, v_sub, v_mul, v_max, v_log_f32, v_mul(neg), v_exp_f32, v_add.

#define SEQ  128
#define COLS 128   // threads per block == batch columns per block

// Hardware transcendentals (base-2): v_log_f32 / v_exp_f32
__device__ __forceinline__ float hw_log2(float x) { return __builtin_amdgcn_logf(x); }
__device__ __forceinline__ float hw_exp2(float x) { return __builtin_amdgcn_exp2f(x); }

__global__ __launch_bounds__(COLS, 1)
void actr_fwd_kernel(const float* __restrict__ sp,
                     const float* __restrict__ w,
                     float* __restrict__ out,
                     int B) {
    // .x = raw sp value, .y = decay_j = c*S_j + a   (128 KB)
    __shared__ float2 tab[SEQ * COLS];

    const int tid = threadIdx.x;
    const int col = blockIdx.x * COLS + tid;
    if (col >= B) return;

    // ---- Stage this thread's column sp[j*B + col] into tab[j*COLS+tid].x
    //      via gfx1250 async global->LDS loads (tracked by ASYNCcnt). ----
    {
        const uint64_t gbase   = (uint64_t)(uintptr_t)sp;
        uint32_t       goff    = (uint32_t)(col * 4);              // byte offset, < 8 MB
        uint32_t       ldsa    = (uint32_t)(uintptr_t)(&tab[tid]); // LDS byte addr of row 0 slot
        const uint32_t gstride = (uint32_t)(B * 4);
        const uint32_t lstride = (uint32_t)(COLS * sizeof(float2));
        for (int j = 0; j < SEQ; ++j) {
            // GVS mode: mem_addr = SADDR64 + VADDR32 ; dsaddr = VDST vgpr
            asm volatile("global_load_async_to_lds_b32 %0, %1, %2"
                         :: "v"(ldsa), "v"(goff), "s"(gbase)
                         : "memory");
            goff += gstride;
            ldsa += lstride;
        }
        asm volatile("s_wait_asynccnt 0x0" ::: "memory");
    }
    // Each thread only touches its own column's LDS cells -> no barrier needed.

    const float a     = w[0];
    const float c_    = w[1];
    const float s_    = w[2];
    const float tau   = w[3];
    const float h     = w[4];
    const float scale = 86400.0f * h;
    const float rinv  = 1.0f / s_;                                  // 1/s
    const float LOG2E = 1.44269504088896340736f;
    const float K     = hw_exp2(tau * rinv * LOG2E);                // exp(tau/s)

    float2* mycol = &tab[tid];                // row stride = COLS float2's
    mycol[0].y = a;                           // m[0] = -inf -> S_0 = 0 -> decay_0 = a

    for (int i = 1; i < SEQ; ++i) {
        const float spi = mycol[i * COLS].x;
        float sum = 0.0f;
        #pragma unroll 4
        for (int j = 0; j < i; ++j) {
            const float2 v = mycol[j * COLS];                       // ds_load_b64 {sp_j, decay_j}
            const float  t = fmaxf((spi - v.x) * scale, 1.0f);
            sum += hw_exp2(-v.y * hw_log2(t));                      // t^(-decay_j)
        }
        // decay_i = c*exp(m_i) + a = c*S_i + a  (exp(ln S) == S: no transcendental)
        mycol[i * COLS].y = c_ * sum + a;
        // out = 1/(1 + exp((tau - ln S)/s)) = 1/(1 + K * S^(-1/s))
        const float e = K * hw_exp2(-rinv * hw_log2(sum));
        out[(i - 1) * B + col] = 1.0f / (1.0f + e);
    }
}

extern "C" void kernel_launch(void* const* d_in, const int* in_sizes, int n_in,
                              void* d_out, int out_size, void* d_ws, size_t ws_size,
                              hipStream_t stream) {
    const float* sp  = (const float*)d_in[0];
    const float* w   = (const float*)d_in[1];
    float*       out = (float*)d_out;
    const int B = in_sizes[0] / SEQ;          // 16384

    dim3 block(COLS);
    dim3 grid((B + COLS - 1) / COLS);         // 128 blocks of 4 waves
    actr_fwd_kernel<<<grid, block, 0, stream>>>(sp, w, out, B);
}